// MaskingGCN_74904229642870
// MI455X (gfx1250) — compile-verified
//
#include <hip/hip_runtime.h>
#include <hip/hip_bf16.h>

// ---------------------------------------------------------------------------
// MaskingGCN on MI455X (gfx1250, wave32).
//   N=100000 nodes, E=3.2M edges, F_IN=4, HID=32, 17 rounds.
// Strategy:
//   * Build CSR (by dst) ONCE per launch: histogram -> exclusive scan ->
//     scatter int32 col indices. Then each of the 16 hidden layers does a
//     zero-atomic GATHER aggregation: one wave per node, 32 lanes = 32
//     features, coalesced 128B reads of L2-resident feature rows.
//   * Combine pass uses v_wmma_f32_16x16x4_f32 (CDNA5 f32 WMMA): one wave
//     per 16-node tile, avg@Wa + h@Wb accumulated K=4 at a time into two
//     16x16 C fragments; row-softmax done in the C fragments via wave32
//     butterfly shuffles; column-softmax exp + column sums fused in.
// ---------------------------------------------------------------------------

typedef __attribute__((ext_vector_type(2))) float v2f;
typedef __attribute__((ext_vector_type(8))) float v8f;

#define N_NODES 100000
#define N_EDGES 3200000
#define HID 32
#define F_IN 4
#define N_HIDDEN 16

// ---------------- CSR build: histogram --------------------------------------
__global__ void hist_kernel(const long long* __restrict__ dst,
                            int* __restrict__ cnt, int nEdges) {
    int t = blockIdx.x * blockDim.x + threadIdx.x;
    if (t < nEdges) atomicAdd(&cnt[(int)dst[t]], 1);
}

__global__ void invdeg_kernel(const int* __restrict__ cnt,
                              float* __restrict__ invd, int n) {
    int t = blockIdx.x * blockDim.x + threadIdx.x;
    if (t < n) invd[t] = 1.0f / fmaxf((float)cnt[t], 1.0f);
}

// ---------------- CSR build: 3-phase exclusive scan -------------------------
__global__ void block_scan_kernel(const int* __restrict__ cnt,
                                  int* __restrict__ rowptr,
                                  int* __restrict__ blockSums, int n) {
    __shared__ int sm[256];
    int t = blockIdx.x * 256 + threadIdx.x;
    int v = (t < n) ? cnt[t] : 0;
    sm[threadIdx.x] = v;
    __syncthreads();
#pragma unroll
    for (int o = 1; o < 256; o <<= 1) {
        int u = (threadIdx.x >= o) ? sm[threadIdx.x - o] : 0;
        __syncthreads();
        sm[threadIdx.x] += u;
        __syncthreads();
    }
    int incl = sm[threadIdx.x];
    if (t < n) rowptr[t] = incl - v;  // exclusive scan within block
    if (threadIdx.x == 255) blockSums[blockIdx.x] = incl;
}

__global__ void scan_sums_kernel(int* __restrict__ blockSums, int nb) {
    if (blockIdx.x == 0 && threadIdx.x == 0) {
        int acc = 0;
        for (int i = 0; i < nb; ++i) {
            int t = blockSums[i];
            blockSums[i] = acc;
            acc += t;
        }
    }
}

__global__ void add_offsets_kernel(int* __restrict__ rowptr,
                                   const int* __restrict__ blockSums, int n) {
    int t = blockIdx.x * 256 + threadIdx.x;
    if (t < n) rowptr[t] += blockSums[blockIdx.x];
}

// ---------------- CSR build: scatter column indices -------------------------
__global__ void csr_scatter_kernel(const long long* __restrict__ src,
                                   const long long* __restrict__ dst,
                                   int* __restrict__ cursor,
                                   int* __restrict__ colidx, int nEdges) {
    int t = blockIdx.x * blockDim.x + threadIdx.x;
    if (t >= nEdges) return;
    int d = (int)dst[t];
    int p = atomicAdd(&cursor[d], 1);
    colidx[p] = (int)src[t];
}

// ---------------- one-shot atomic aggregation for proj_in (F_IN=4) ----------
__global__ void agg4_kernel(const long long* __restrict__ src,
                            const long long* __restrict__ dst,
                            const float* __restrict__ x,
                            float* __restrict__ msg, int nEdges) {
    int t = blockIdx.x * blockDim.x + threadIdx.x;
    if (t >= nEdges) return;
    int s = (int)src[t];
    int d = (int)dst[t];
    const float4 v = *(const float4*)&x[(long long)s * F_IN];
    float* m = &msg[(long long)d * F_IN];
    atomicAdd(m + 0, v.x);
    atomicAdd(m + 1, v.y);
    atomicAdd(m + 2, v.z);
    atomicAdd(m + 3, v.w);
}

// ---------------- per-layer gather aggregation (zero atomics) ---------------
// One wave per node: lane = feature column. Neighbor index is wave-uniform
// (scalarizable); h-row reads coalesce to 128B. 4-way unrolled for MLP.
__global__ void gather_kernel(const int* __restrict__ rowptr,
                              const int* __restrict__ cnt,
                              const int* __restrict__ colidx,
                              const float* __restrict__ h,
                              float* __restrict__ msg, int n) {
    int lane = threadIdx.x & 31;
    int node = blockIdx.x * (blockDim.x >> 5) + (threadIdx.x >> 5);
    if (node >= n) return;
    int beg = rowptr[node];
    int end = beg + cnt[node];
    float acc = 0.f;
    int j = beg;
    for (; j + 4 <= end; j += 4) {
        int n0 = colidx[j + 0];
        int n1 = colidx[j + 1];
        int n2 = colidx[j + 2];
        int n3 = colidx[j + 3];
        float a0 = h[(long long)n0 * HID + lane];
        float a1 = h[(long long)n1 * HID + lane];
        float a2 = h[(long long)n2 * HID + lane];
        float a3 = h[(long long)n3 * HID + lane];
        acc += a0 + a1 + a2 + a3;
    }
    for (; j < end; ++j) acc += h[(long long)colidx[j] * HID + lane];
    msg[(long long)node * HID + lane] = acc;
}

// ---------------- combine: h' = rowsoftmax(avg@Wa + h@Wb) -------------------
// One wave handles a 16-row tile, full 32 output columns (two 16-col C tiles).
// A-frag (16x4 f32): lanes 0-15 hold (row, K=k0..k0+1), lanes 16-31 hold
// (row, K=k0+2..k0+3).  B-frag (4x16): K rows striped across lanes.
// C (16x16 f32): VGPR r <-> row r (lanes 0-15) / r+8 (lanes 16-31).
// N_NODES % 16 == 0, so active waves always have full EXEC (WMMA req);
// tail waves exit wave-uniformly before any WMMA.
template <int KDIM, bool COLSOFT>
__global__ void combine_kernel(const float* __restrict__ msg,   // [n x KDIM]
                               const float* __restrict__ hin,   // [n x KDIM]
                               const float* __restrict__ invd,  // [n]
                               const float* __restrict__ Wa,    // [KDIM x 32]
                               const float* __restrict__ Wb,    // [KDIM x 32]
                               float* __restrict__ hout,        // [n x 32]
                               float* __restrict__ colsum,      // [32]
                               int nrows) {
    const int lane = threadIdx.x & 31;
    const int wave = threadIdx.x >> 5;
    const int tile = blockIdx.x * (blockDim.x >> 5) + wave;
    if (tile * 16 >= nrows) return;  // wave-uniform exit
    const int half = lane >> 4;
    const int lr = lane & 15;
    const int arow = tile * 16 + lr;
    const float id = invd[arow];

    v8f c0 = {};  // columns 0..15
    v8f c1 = {};  // columns 16..31
#pragma unroll
    for (int k0 = 0; k0 < KDIM; k0 += 4) {
        const int ka = k0 + 2 * half;
        v2f am = *(const v2f*)&msg[(long long)arow * KDIM + ka];
        am *= id;  // mean aggregation
        v2f ah = *(const v2f*)&hin[(long long)arow * KDIM + ka];
        v2f ba0 = {Wa[ka * 32 + lr], Wa[(ka + 1) * 32 + lr]};
        v2f ba1 = {Wa[ka * 32 + 16 + lr], Wa[(ka + 1) * 32 + 16 + lr]};
        v2f bb0 = {Wb[ka * 32 + lr], Wb[(ka + 1) * 32 + lr]};
        v2f bb1 = {Wb[ka * 32 + 16 + lr], Wb[(ka + 1) * 32 + 16 + lr]};
        c0 = __builtin_amdgcn_wmma_f32_16x16x4_f32(false, am, false, ba0,
                                                   (short)0, c0, false, false);
        c1 = __builtin_amdgcn_wmma_f32_16x16x4_f32(false, am, false, ba1,
                                                   (short)0, c1, false, false);
        c0 = __builtin_amdgcn_wmma_f32_16x16x4_f32(false, ah, false, bb0,
                                                   (short)0, c0, false, false);
        c1 = __builtin_amdgcn_wmma_f32_16x16x4_f32(false, ah, false, bb1,
                                                   (short)0, c1, false, false);
    }

    // Row softmax in-registers: row r's 32 features live in c0[r], c1[r]
    // across 16 lanes of one wave half (xor masks 1,2,4,8 stay in the half).
    float colacc0 = 0.f, colacc1 = 0.f;
#pragma unroll
    for (int r = 0; r < 8; ++r) {
        float u0 = c0[r], u1 = c1[r];
        float m = fmaxf(u0, u1);
#pragma unroll
        for (int msk = 1; msk <= 8; msk <<= 1)
            m = fmaxf(m, __shfl_xor(m, msk, 32));
        float e0 = __expf(u0 - m);
        float e1 = __expf(u1 - m);
        float s = e0 + e1;
#pragma unroll
        for (int msk = 1; msk <= 8; msk <<= 1) s += __shfl_xor(s, msk, 32);
        float rinv = 1.0f / s;
        float h0 = e0 * rinv;
        float h1 = e1 * rinv;
        const int orow = tile * 16 + r + 8 * half;
        if (COLSOFT) {
            // next step is softmax over nodes per column; inputs are in (0,1)
            // so exp() without max-subtraction is numerically safe.
            h0 = __expf(h0);
            h1 = __expf(h1);
            colacc0 += h0;
            colacc1 += h1;
        }
        hout[(long long)orow * 32 + lr] = h0;
        hout[(long long)orow * 32 + 16 + lr] = h1;
    }
    if (COLSOFT) {
        colacc0 += __shfl_xor(colacc0, 16, 32);  // fold wave halves
        colacc1 += __shfl_xor(colacc1, 16, 32);
        if (lane < 16) {
            atomicAdd(&colsum[lr], colacc0);
            atomicAdd(&colsum[16 + lr], colacc1);
        }
    }
}

// ---------------- column-softmax normalization ------------------------------
__global__ void colnorm_kernel(float* __restrict__ h,
                               const float* __restrict__ colsum, int n4) {
    int t = blockIdx.x * blockDim.x + threadIdx.x;
    if (t >= n4) return;
    int base = t * 4;
    int cb = base & 31;  // 32 % 4 == 0 -> float4 stays in-row
    float4 v = *(const float4*)&h[base];
    v.x /= colsum[cb + 0];
    v.y /= colsum[cb + 1];
    v.z /= colsum[cb + 2];
    v.w /= colsum[cb + 3];
    *(float4*)&h[base] = v;
}

// ---------------- output head: exp(h@Wout + b), wave-reduced sum ------------
__global__ void out_kernel(const float* __restrict__ h,
                           const float* __restrict__ Wout,
                           const float* __restrict__ bout,
                           float* __restrict__ out,
                           float* __restrict__ sumbuf, int n) {
    int i = blockIdx.x * blockDim.x + threadIdx.x;
    float e = 0.f;
    if (i < n) {
        float z = bout[0];
#pragma unroll
        for (int j = 0; j < 32; j += 4) {
            float4 v = *(const float4*)&h[(long long)i * 32 + j];
            float4 w = *(const float4*)&Wout[j];
            z += v.x * w.x + v.y * w.y + v.z * w.z + v.w * w.w;
        }
        e = __expf(z);  // values are tiny; no max pass needed
        out[i] = e;
    }
#pragma unroll
    for (int msk = 1; msk < 32; msk <<= 1) e += __shfl_xor(e, msk, 32);
    if ((threadIdx.x & 31) == 0) atomicAdd(sumbuf, e);
}

__global__ void outnorm_kernel(float* __restrict__ out,
                               const float* __restrict__ sumbuf, int n) {
    int i = blockIdx.x * blockDim.x + threadIdx.x;
    if (i < n) out[i] /= sumbuf[0];
}

// ---------------------------------------------------------------------------
static inline int cdiv(long long a, int b) { return (int)((a + b - 1) / b); }

extern "C" void kernel_launch(void* const* d_in, const int* in_sizes, int n_in,
                              void* d_out, int out_size, void* d_ws,
                              size_t ws_size, hipStream_t stream) {
    const float* x = (const float*)d_in[0];           // [N,4]
    const long long* ei = (const long long*)d_in[1];  // [2,E] int64
    const float* A_in = (const float*)d_in[2];        // [4,32]
    const float* B_in = (const float*)d_in[3];        // [4,32]
    const float* A_conv = (const float*)d_in[4];      // [16,32,32]
    const float* B_conv = (const float*)d_in[5];      // [16,32,32]
    const float* W_out = (const float*)d_in[6];       // [32,1]
    const float* b_out = (const float*)d_in[7];       // [1]
    float* out = (float*)d_out;                       // [N]

    const long long* src = ei;
    const long long* dst = ei + N_EDGES;

    // ---- workspace layout (16B aligned pieces) ----
    char* ws = (char*)d_ws;
    size_t off = 0;
    float* invd = (float*)(ws + off);   off += (size_t)N_NODES * 4;
    int* cnt    = (int*)(ws + off);     off += (size_t)N_NODES * 4;
    int* rowptr = (int*)(ws + off);     off += (size_t)N_NODES * 4;
    int* cursor = (int*)(ws + off);     off += (size_t)N_NODES * 4;
    int* colidx = (int*)(ws + off);     off += (size_t)N_EDGES * 4;
    float* msg  = (float*)(ws + off);   off += (size_t)N_NODES * HID * 4;
    float* hA   = (float*)(ws + off);   off += (size_t)N_NODES * HID * 4;
    float* hB   = (float*)(ws + off);   off += (size_t)N_NODES * HID * 4;
    int* blockSums = (int*)(ws + off);  off += 1024 * 4;
    float* colsum  = (float*)(ws + off); off += 64 * 4;
    float* sumbuf  = (float*)(ws + off); off += 16 * 4;
    (void)ws_size; (void)in_sizes; (void)n_in; (void)out_size;

    const int TPB = 256;
    const int scanBlocks = cdiv(N_NODES, 256);  // 391

    // ---- CSR build (once per launch, amortized over 16 gather layers) ----
    hipMemsetAsync(cnt, 0, (size_t)N_NODES * 4, stream);
    hist_kernel<<<cdiv(N_EDGES, TPB), TPB, 0, stream>>>(dst, cnt, N_EDGES);
    invdeg_kernel<<<cdiv(N_NODES, TPB), TPB, 0, stream>>>(cnt, invd, N_NODES);
    block_scan_kernel<<<scanBlocks, 256, 0, stream>>>(cnt, rowptr, blockSums,
                                                      N_NODES);
    scan_sums_kernel<<<1, 64, 0, stream>>>(blockSums, scanBlocks);
    add_offsets_kernel<<<scanBlocks, 256, 0, stream>>>(rowptr, blockSums,
                                                       N_NODES);
    hipMemcpyAsync(cursor, rowptr, (size_t)N_NODES * 4,
                   hipMemcpyDeviceToDevice, stream);
    csr_scatter_kernel<<<cdiv(N_EDGES, TPB), TPB, 0, stream>>>(
        src, dst, cursor, colidx, N_EDGES);

    // ---- proj_in: h = rowsoftmax(avg(x)@A_in + x@B_in) ----
    hipMemsetAsync(msg, 0, (size_t)N_NODES * F_IN * 4, stream);
    agg4_kernel<<<cdiv(N_EDGES, TPB), TPB, 0, stream>>>(src, dst, x, msg,
                                                        N_EDGES);
    {
        const int tiles = N_NODES / 16;     // 6250, exact
        const int blocks = cdiv(tiles, 4);  // 4 waves per 128-thread block
        combine_kernel<F_IN, false><<<blocks, 128, 0, stream>>>(
            msg, x, invd, A_in, B_in, hA, nullptr, N_NODES);
    }

    // ---- 16 hidden conv layers: gather (no atomics) + WMMA combine ----
    float* cur = hA;
    float* nxt = hB;
    for (int l = 0; l < N_HIDDEN; ++l) {
        hipMemsetAsync(colsum, 0, 32 * 4, stream);
        gather_kernel<<<cdiv(N_NODES, 8), 256, 0, stream>>>(
            rowptr, cnt, colidx, cur, msg, N_NODES);
        const int tiles = N_NODES / 16;
        const int blocks = cdiv(tiles, 4);
        combine_kernel<HID, true><<<blocks, 128, 0, stream>>>(
            msg, cur, invd, A_conv + (size_t)l * HID * HID,
            B_conv + (size_t)l * HID * HID, nxt, colsum, N_NODES);
        colnorm_kernel<<<cdiv((long long)N_NODES * (HID / 4), TPB), TPB, 0,
                         stream>>>(nxt, colsum, N_NODES * (HID / 4));
        float* t = cur; cur = nxt; nxt = t;
    }

    // ---- output head: colsoftmax(h @ W_out + b_out) ----
    hipMemsetAsync(sumbuf, 0, 4, stream);
    out_kernel<<<cdiv(N_NODES, TPB), TPB, 0, stream>>>(cur, W_out, b_out, out,
                                                       sumbuf, N_NODES);
    outnorm_kernel<<<cdiv(N_NODES, TPB), TPB, 0, stream>>>(out, sumbuf,
                                                           N_NODES);
}